// PyramidROIAlign_21792664060113
// MI455X (gfx1250) — compile-verified
//
#include <hip/hip_runtime.h>

typedef __attribute__((ext_vector_type(2))) float v2f;
typedef __attribute__((ext_vector_type(8))) float v8f;

namespace {
constexpr int kBatch = 2;
constexpr int kBoxes = 1000;
constexpr int kCh    = 256;
constexpr int kPH    = 7;
constexpr int kPW    = 7;
}

// ---------------------------------------------------------------------------
// Stable counting-rank per batch: rank(n) = #{m : lvl[m] < lvl[n]}
//                                        + #{m < n : lvl[m] == lvl[n]}
// Scatter src_idx[b][rank] = n  ==  argsort(level, stable) per batch.
// ---------------------------------------------------------------------------
__global__ __launch_bounds__(1024) void rank_kernel(const int* __restrict__ level,
                                                    int* __restrict__ src_idx) {
  __shared__ int lv[kBoxes];
  const int b = blockIdx.x;
  for (int i = threadIdx.x; i < kBoxes; i += blockDim.x) lv[i] = level[b * kBoxes + i];
  __syncthreads();
  for (int n = threadIdx.x; n < kBoxes; n += blockDim.x) {
    const int myl = lv[n];
    int rank = 0;
    for (int m = 0; m < kBoxes; ++m) {
      const int l = lv[m];
      rank += (int)((l < myl) || ((l == myl) && (m < n)));
    }
    src_idx[b * kBoxes + rank] = n;  // ranks are a permutation: conflict-free
  }
}

// ---------------------------------------------------------------------------
// One wave32 per (batch, sorted-slot, gy, gx) cell.
// Bilinear combine D = A(16x4 weights) x B(4x16 corner values) via
// V_WMMA_F32_16X16X4_F32. Channel groups processed in pairs: result of the
// even group sits in lanes 0-15 (M=0 row), result of the odd group is
// duplicated into lanes 16-31 (M=8 row == same channels), so one cndmask
// merge gives a full-wave 128B contiguous store with no EXEC masking.
// ---------------------------------------------------------------------------
__global__ __launch_bounds__(256) void pool_kernel(
    const float* __restrict__ boxes,
    const float* __restrict__ fm0, const float* __restrict__ fm1,
    const float* __restrict__ fm2, const float* __restrict__ fm3,
    const int* __restrict__ level, const int* __restrict__ src_idx,
    float* __restrict__ out) {
  const int lane = (int)(threadIdx.x & 31u);
  const int wave = (int)blockIdx.x * ((int)blockDim.x >> 5) + ((int)threadIdx.x >> 5);

  const int cell = wave % (kPH * kPW);
  const int bp   = wave / (kPH * kPW);   // b * kBoxes + sorted output position
  const int gx = cell % kPW;
  const int gy = cell / kPW;
  const int b  = bp / kBoxes;

  const int n  = src_idx[bp];            // source box for this sorted slot
  const int bn = b * kBoxes + n;

  const float y1 = boxes[bn * 4 + 0];
  const float x1 = boxes[bn * 4 + 1];
  const float y2 = boxes[bn * 4 + 2];
  const float x2 = boxes[bn * 4 + 3];

  const int lvl = level[bn];             // 1..4
  const float* fm;
  int S;
  if (lvl == 1)      { fm = fm0; S = 256; }
  else if (lvl == 2) { fm = fm1; S = 128; }
  else if (lvl == 3) { fm = fm2; S = 64;  }
  else               { fm = fm3; S = 32;  }

  const float Sm1 = (float)(S - 1);
  const float in_y = y1 * Sm1 + (float)gy * ((y2 - y1) * Sm1 * (1.0f / (kPH - 1)));
  const float in_x = x1 * Sm1 + (float)gx * ((x2 - x1) * Sm1 * (1.0f / (kPW - 1)));

  const bool valid = (in_y >= 0.0f) && (in_y <= Sm1) && (in_x >= 0.0f) && (in_x <= Sm1);

  const float fy = floorf(in_y);
  const float fx = floorf(in_x);
  const int ty = (int)fminf(fmaxf(fy, 0.0f), Sm1);
  const int lx = (int)fminf(fmaxf(fx, 0.0f), Sm1);
  const int by = min(ty + 1, S - 1);
  const int rx = min(lx + 1, S - 1);
  const float wy = in_y - fy;
  const float wx = in_x - fx;

  // A-matrix layout (16x4 f32, K: 0=tl 1=tr 2=bl 3=br):
  //   VGPR0: lanes 0-15 -> K=0, lanes 16-31 -> K=2
  //   VGPR1: lanes 0-15 -> K=1, lanes 16-31 -> K=3
  const bool hi = lane >= 16;
  const float vmask = valid ? 1.0f : 0.0f;         // extrapolation_value = 0
  const float wrow  = (hi ? wy : (1.0f - wy)) * vmask;
  v2f A;
  A.x = wrow * (1.0f - wx);
  A.y = wrow * wx;

  // B-matrix layout (4x16 f32): VGPR0 rows K0/K2, VGPR1 rows K1/K3,
  // column N = lane&15 = channel within the 16-channel group.
  const int row = hi ? by : ty;
  const int ch  = lane & 15;
  const float* rbase = fm + ((size_t)b * (size_t)S + (size_t)row) * (size_t)S * kCh;
  const float* p0 = rbase + (size_t)lx * kCh + ch;  // K0 / K2 column (left)
  const float* p1 = rbase + (size_t)rx * kCh + ch;  // K1 / K3 column (right)

  // Full-wave output pointer: channel == lane for each 32-channel pair.
  float* outp = out + ((((size_t)bp * kPH + gy) * kPW + gx) * kCh) + lane;

  // Batch-preload all 32 B operands (one clause, latency fully overlapped).
  float b0[16], b1[16];
#pragma unroll
  for (int j = 0; j < 16; ++j) {
    b0[j] = p0[j * 16];
    b1[j] = p1[j * 16];
  }

#pragma unroll
  for (int g = 0; g < 8; ++g) {
    v2f Be; Be.x = b0[2 * g];     Be.y = b1[2 * g];      // channels 32g..32g+15
    v2f Bo; Bo.x = b0[2 * g + 1]; Bo.y = b1[2 * g + 1];  // channels 32g+16..32g+31
    v8f acce = {};
    v8f acco = {};
    acce = __builtin_amdgcn_wmma_f32_16x16x4_f32(
        false, A, false, Be, (short)0, acce, false, false);
    acco = __builtin_amdgcn_wmma_f32_16x16x4_f32(
        false, A, false, Bo, (short)0, acco, false, false);
    // lanes 0-15: even group (M=0 row); lanes 16-31: odd group (M=8 row dup).
    const float res = hi ? acco[0] : acce[0];
    // NT store: output is write-once; keep the feature maps resident in L2.
    __builtin_nontemporal_store(res, outp + g * 32);
  }
}

extern "C" void kernel_launch(void* const* d_in, const int* in_sizes, int n_in,
                              void* d_out, int out_size, void* d_ws, size_t ws_size,
                              hipStream_t stream) {
  const float* boxes = (const float*)d_in[0];
  const float* fm0   = (const float*)d_in[1];
  const float* fm1   = (const float*)d_in[2];
  const float* fm2   = (const float*)d_in[3];
  const float* fm3   = (const float*)d_in[4];
  const int*   level = (const int*)d_in[5];

  int*   src_idx = (int*)d_ws;           // kBatch * kBoxes ints
  float* out     = (float*)d_out;

  rank_kernel<<<kBatch, 1024, 0, stream>>>(level, src_idx);

  const int total_waves     = kBatch * kBoxes * kPH * kPW;  // 98000
  const int waves_per_block = 256 / 32;                     // 8
  pool_kernel<<<total_waves / waves_per_block, 256, 0, stream>>>(
      boxes, fm0, fm1, fm2, fm3, level, src_idx, out);
}